// SparseRegression_85048942395528
// MI455X (gfx1250) — compile-verified
//
#include <hip/hip_runtime.h>
#include <cstdint>

// ---------------------------------------------------------------------------
// Top-2 disparity regression over C=24 channels of an 8x24x320x640 volume.
// Memory-bound (~335 MB @ 23.3 TB/s ~= 14 us). Data movement uses the CDNA5
// Tensor Data Mover: one 2D-tile DMA (24 rows x 1KB, row stride H*W*4 bytes)
// per input array per block, Global -> LDS, tracked with TENSORcnt.
// Outputs are written with non-temporal hints (write-once, never re-read).
// ---------------------------------------------------------------------------

typedef __attribute__((ext_vector_type(4))) unsigned int u32x4;
typedef __attribute__((ext_vector_type(8))) int          i32x8;
typedef __attribute__((ext_vector_type(4))) int          i32x4;

constexpr int kB    = 8;
constexpr int kC    = 24;
constexpr int kH    = 320;
constexpr int kW    = 640;
constexpr int kHW   = kH * kW;          // 204800 (divisible by kTile)
constexpr int kTile = 256;              // pixels per block

// Issue a TDM load of a 2D tile: kTile contiguous fp32 elements (dim0) x kC
// rows (dim1), row stride kHW elements, Global -> LDS (row-major [kC][kTile]).
// D# bit layout per CDNA5 ISA 8.3/8.4. Groups 2/3 zero (2D tile: dims 2..4
// unused, iterate/gather/pad disabled). 6-arg builtin on this toolchain:
// (g0, g1, g2, g3, extra int32x8, cpol).
__device__ __forceinline__ void tdm_load_tile(uint32_t lds_byte_addr,
                                              const float* gptr) {
  u32x4 g0;
  uint64_t ga = (uint64_t)(uintptr_t)gptr;
  g0[0] = 1u;                                   // [1:0] count=1 (user, valid)
  g0[1] = lds_byte_addr;                        // [63:32] lds_addr (bytes)
  g0[2] = (uint32_t)ga;                         // [95:64] global_addr lo
  g0[3] = (uint32_t)(ga >> 32) | (2u << 30);    // [120:96] addr hi, [127:126]=2

  i32x8 g1;
  g1[0] = 0x00020000;                           // [17:16] data_size=2 (4 bytes)
  g1[1] = (int)((kHW & 0xFFFF) << 16);          // [63:48] tensor_dim0 lo16
  g1[2] = (int)(((unsigned)kHW >> 16) |         // [79:64] tensor_dim0 hi16
                ((unsigned)kC << 16));          // [95:80] tensor_dim1 lo16
  g1[3] = (int)((unsigned)kTile << 16);         // [127:112] tile_dim0 = 256
  g1[4] = kC;                                   // [143:128] tile_dim1 = 24
  g1[5] = kHW;                                  // [191:160] dim0_stride lo32
  g1[6] = 0;                                    // stride hi / dim1_stride lo
  g1[7] = 0;

  i32x4 z4 = {0, 0, 0, 0};
  i32x8 z8 = {0, 0, 0, 0, 0, 0, 0, 0};
  __builtin_amdgcn_tensor_load_to_lds(g0, g1, z4, z4, z8, 0);
}

__global__ __launch_bounds__(kTile) void
topk2_regress(const float* __restrict__ cost,
              const float* __restrict__ disp,
              float* __restrict__ pred,     // [B, H, W]
              float* __restrict__ prob) {   // [B, 2, H, W]
  __shared__ float s_cost[kC * kTile];      // 24 KB
  __shared__ float s_disp[kC * kTile];      // 24 KB

  const int tid  = threadIdx.x;
  const int tile = blockIdx.x;
  const long long p0idx = (long long)tile * kTile;  // first flat pixel index
  const int b   = (int)(p0idx / kHW);
  const int hw0 = (int)(p0idx - (long long)b * kHW);

  const float* gcost = cost + (size_t)b * kC * kHW + hw0;
  const float* gdisp = disp + (size_t)b * kC * kHW + hw0;

  // One TDM per input array; waves 0 and 1 each own one descriptor so the two
  // DMAs issue independently. Low 32 bits of a flat LDS pointer == LDS offset.
  const int wave = tid >> 5;
  if (wave == 0) {
    tdm_load_tile((uint32_t)(uintptr_t)&s_cost[0], gcost);
  } else if (wave == 1) {
    tdm_load_tile((uint32_t)(uintptr_t)&s_disp[0], gdisp);
  }
  __builtin_amdgcn_s_wait_tensorcnt(0);     // issuing waves: DMA complete
  __syncthreads();                          // publish LDS to all waves

  // Per-pixel top-2 scan (first occurrence wins ties, matching lax.top_k).
  float m0 = -INFINITY, m1 = -INFINITY;
  float d0 = 0.0f, d1 = 0.0f;
#pragma unroll
  for (int c = 0; c < kC; ++c) {
    const float cv = s_cost[c * kTile + tid];
    const float dv = s_disp[c * kTile + tid];
    if (cv > m0) {
      m1 = m0; d1 = d0;
      m0 = cv; d0 = dv;
    } else if (cv > m1) {
      m1 = cv; d1 = dv;
    }
  }

  // softmax over {m0, m1} with m0 = max
  const float e1 = expf(m1 - m0);
  const float pr0 = 1.0f / (1.0f + e1);
  const float pr1 = e1 * pr0;

  // Coalesced, non-temporal stores (outputs are write-once).
  __builtin_nontemporal_store(d0 * pr0 + d1 * pr1, &pred[p0idx + tid]);
  float* pbase = prob + (size_t)b * 2 * kHW + hw0 + tid;
  __builtin_nontemporal_store(pr0, &pbase[0]);     // K = 0 plane
  __builtin_nontemporal_store(pr1, &pbase[kHW]);   // K = 1 plane
}

extern "C" void kernel_launch(void* const* d_in, const int* in_sizes, int n_in,
                              void* d_out, int out_size, void* d_ws,
                              size_t ws_size, hipStream_t stream) {
  (void)in_sizes; (void)n_in; (void)out_size; (void)d_ws; (void)ws_size;
  const float* cost = (const float*)d_in[0];
  const float* disp = (const float*)d_in[1];
  float* pred = (float*)d_out;                    // B*H*W floats
  float* prob = (float*)d_out + (size_t)kB * kHW; // then B*2*H*W floats

  const int tiles = kB * kHW / kTile;             // 6400 blocks
  topk2_regress<<<dim3(tiles), dim3(kTile), 0, stream>>>(cost, disp, pred, prob);
}